// Qmixer_64896955842839
// MI455X (gfx1250) — compile-verified
//
#include <hip/hip_runtime.h>
#include <hip/hip_bf16.h>
#include <math.h>

// ---------------------------------------------------------------------------
// QMixer: one workgroup per graph (2048 graphs x 128 nodes, 64 allies, D=64,
// K=10). All per-graph dense math done with V_WMMA_F32_16X16X4_F32 from LDS.
// ---------------------------------------------------------------------------

typedef __attribute__((ext_vector_type(2))) float v2f;
typedef __attribute__((ext_vector_type(8))) float v8f;

#define NF_STRIDE 68   // 64 + pad, rows stay 16B aligned (68*4 = 272 = 17*16)
#define WF_STRIDE 68

__global__ void qmixer_zero_kernel(float* __restrict__ p, int n) {
    int i = blockIdx.x * blockDim.x + threadIdx.x;
    int stride = gridDim.x * blockDim.x;
    for (; i < n; i += stride) p[i] = 0.0f;
}

__global__ __launch_bounds__(128) void qmixer_graph_kernel(
    const float* __restrict__ node_feature,   // [N, 64]
    const float* __restrict__ qs,             // [NA]
    const float* __restrict__ Ww,             // [64, 10]
    const float* __restrict__ bw,             // [10]
    const float* __restrict__ W1,             // [64, 64]
    const float* __restrict__ b1,             // [64]
    const float* __restrict__ W2,             // [64, 10]
    const float* __restrict__ b2,             // [10]
    const int*   __restrict__ ally_indices,   // [NA]
    float* __restrict__ out_qagg,             // [B, 10]
    float* __restrict__ out_ws,               // [NA, 10]
    float* __restrict__ out_wf,               // [B, 10, 64]
    float* __restrict__ out_normed,           // [NA, 10]
    float* __restrict__ out_full)             // [N, 10]
{
    __shared__ float s_nf[64 * NF_STRIDE];    // ally node features [64][64]
    __shared__ float s_Ww[64 * 16];           // Ww padded to 16 clusters
    __shared__ float s_ws[64 * 16];           // scores -> softmaxed ws
    __shared__ float s_wf[16 * WF_STRIDE];    // wf [16][64]
    __shared__ float s_dot[64 * 16];          // group_dot
    __shared__ float s_qs[64];
    __shared__ float s_wfnorm[16];
    __shared__ float s_part[2 * 64];
    __shared__ float s_sumnf[64];
    __shared__ float s_hidden[64];
    __shared__ float s_qsum[16];
    __shared__ int   s_aidx[64];

    const int g    = blockIdx.x;
    const int tid  = threadIdx.x;
    const int lane = tid & 31;
    const int wave = tid >> 5;       // 0..3
    const int ln   = lane & 15;
    const int hi   = lane >> 4;      // 0 or 1

    // ---------------- Phase A: stage inputs into LDS ----------------
    if (tid < 64) {
        s_aidx[tid] = ally_indices[g * 64 + tid];
        s_qs[tid]   = qs[g * 64 + tid];
    } else {
        int r = tid - 64;  // 0..63 : one Ww row each, pad clusters 10..15
        #pragma unroll
        for (int k = 0; k < 10; ++k) s_Ww[r * 16 + k] = Ww[r * 10 + k];
        #pragma unroll
        for (int k = 10; k < 16; ++k) s_Ww[r * 16 + k] = 0.0f;
    }
    __syncthreads();

    {   // 128 threads: each loads half an ally feature row (32 floats, float4)
        int row  = tid >> 1;
        int half = tid & 1;
        const float4* src = (const float4*)(node_feature + (size_t)s_aidx[row] * 64 + half * 32);
        float4* dst = (float4*)(s_nf + row * NF_STRIDE + half * 32);
        #pragma unroll
        for (int j = 0; j < 8; ++j) dst[j] = src[j];
    }
    __syncthreads();

    // ---------------- Phase B: scores = allyNF @ Ww + bw (WMMA) ----------------
    // wave w owns ally rows [16w, 16w+16); N = 16 (clusters padded)
    float bwv = (ln < 10) ? bw[ln] : 0.0f;
    v8f acc;
    #pragma unroll
    for (int r = 0; r < 8; ++r) acc[r] = bwv;   // + bias on every row

    #pragma unroll
    for (int k0 = 0; k0 < 64; k0 += 4) {
        int kk = k0 + hi * 2;
        int arow = wave * 16 + ln;
        v2f a, b;
        a.x = s_nf[arow * NF_STRIDE + kk];
        a.y = s_nf[arow * NF_STRIDE + kk + 1];
        b.x = s_Ww[kk * 16 + ln];
        b.y = s_Ww[(kk + 1) * 16 + ln];
        acc = __builtin_amdgcn_wmma_f32_16x16x4_f32(false, a, false, b,
                                                    (short)0, acc, false, false);
    }
    #pragma unroll
    for (int r = 0; r < 8; ++r) {
        int row = wave * 16 + r + hi * 8;
        s_ws[row * 16 + ln] = acc[r];
    }
    __syncthreads();

    // ---------------- Phase C: clip + softmax over K=10 ----------------
    if (tid < 64) {
        float v[10];
        float m = -1e30f;
        #pragma unroll
        for (int k = 0; k < 10; ++k) {
            float x = s_ws[tid * 16 + k];
            x = fminf(fmaxf(x, 1e-10f), 10.0f);
            v[k] = x;
            m = fmaxf(m, x);
        }
        float sum = 0.0f;
        #pragma unroll
        for (int k = 0; k < 10; ++k) { v[k] = __expf(v[k] - m); sum += v[k]; }
        float inv = 1.0f / sum;
        #pragma unroll
        for (int k = 0; k < 10; ++k) {
            float w = v[k] * inv;
            s_ws[tid * 16 + k] = w;
            out_ws[(size_t)(g * 64 + tid) * 10 + k] = w;
        }
        #pragma unroll
        for (int k = 10; k < 16; ++k) s_ws[tid * 16 + k] = 0.0f;
    }
    __syncthreads();

    // ---------------- Phase D: wf = ws^T @ allyNF (WMMA) ----------------
    // M = 16 clusters, wave w owns feature cols [16w, 16w+16), Kdim = 64 allies
    v8f acc2;
    #pragma unroll
    for (int r = 0; r < 8; ++r) acc2[r] = 0.0f;
    #pragma unroll
    for (int k0 = 0; k0 < 64; k0 += 4) {
        int kk = k0 + hi * 2;
        v2f a, b;
        a.x = s_ws[kk * 16 + ln];          // A[m][k] = ws[k][m]
        a.y = s_ws[(kk + 1) * 16 + ln];
        int bcol = wave * 16 + ln;
        b.x = s_nf[kk * NF_STRIDE + bcol]; // B[k][n] = nf[k][n]
        b.y = s_nf[(kk + 1) * NF_STRIDE + bcol];
        acc2 = __builtin_amdgcn_wmma_f32_16x16x4_f32(false, a, false, b,
                                                     (short)0, acc2, false, false);
    }
    #pragma unroll
    for (int r = 0; r < 8; ++r) {
        int m = r + hi * 8;                // cluster
        int d = wave * 16 + ln;            // feature
        s_wf[m * WF_STRIDE + d] = acc2[r];
        if (m < 10) out_wf[((size_t)g * 10 + m) * 64 + d] = acc2[r];
    }
    __syncthreads();

    // ---------------- Phase E: group_dot = allyNF @ wf^T (WMMA) ----------------
    v8f acc3;
    #pragma unroll
    for (int r = 0; r < 8; ++r) acc3[r] = 0.0f;
    #pragma unroll
    for (int k0 = 0; k0 < 64; k0 += 4) {
        int kk = k0 + hi * 2;
        int arow = wave * 16 + ln;
        v2f a, b;
        a.x = s_nf[arow * NF_STRIDE + kk];
        a.y = s_nf[arow * NF_STRIDE + kk + 1];
        b.x = s_wf[ln * WF_STRIDE + kk];   // B[k][n] = wf[n][k]
        b.y = s_wf[ln * WF_STRIDE + kk + 1];
        acc3 = __builtin_amdgcn_wmma_f32_16x16x4_f32(false, a, false, b,
                                                     (short)0, acc3, false, false);
    }
    #pragma unroll
    for (int r = 0; r < 8; ++r) {
        int row = wave * 16 + r + hi * 8;
        s_dot[row * 16 + ln] = acc3[r];
    }
    __syncthreads();

    // ---------------- Phase F: norms + normed outputs ----------------
    if (tid >= 64 && tid < 80) {
        int k = tid - 64;
        float s = 0.0f;
        #pragma unroll 8
        for (int d = 0; d < 64; ++d) { float x = s_wf[k * WF_STRIDE + d]; s += x * x; }
        s_wfnorm[k] = sqrtf(s);
    }
    __syncthreads();
    if (tid < 64) {
        float s = 0.0f;
        #pragma unroll 8
        for (int d = 0; d < 64; ++d) { float x = s_nf[tid * NF_STRIDE + d]; s += x * x; }
        float nfn = sqrtf(s);
        int ai = s_aidx[tid];
        #pragma unroll
        for (int k = 0; k < 10; ++k) {
            float val = s_dot[tid * 16 + k] / (nfn * s_wfnorm[k]);
            out_normed[(size_t)(g * 64 + tid) * 10 + k] = val;
            out_full[(size_t)ai * 10 + k] = val;
        }
    }

    // ---------------- Phase G: q_agg (weighted q) ----------------
    __syncthreads();
    if (tid < 16) {
        float s = 0.0f;
        if (tid < 10) {
            #pragma unroll 8
            for (int r = 0; r < 64; ++r) s += s_qs[r] * s_ws[r * 16 + tid];
        }
        s_qsum[tid] = s;
    }

    // ---------------- Phase H: sum of ALL 128 node features ----------------
    {
        int d = tid & 63;
        int half = tid >> 6;
        const float* base = node_feature + ((size_t)g * 128 + half * 64) * 64;
        float s = 0.0f;
        #pragma unroll 8
        for (int n = 0; n < 64; ++n) s += base[n * 64 + d];
        s_part[half * 64 + d] = s;
    }
    __syncthreads();
    if (tid < 64) s_sumnf[tid] = s_part[tid] + s_part[64 + tid];
    __syncthreads();

    // ---------------- Phase I: q_b_net MLP + final q_agg ----------------
    if (tid < 64) {
        float s = b1[tid];
        #pragma unroll 8
        for (int d = 0; d < 64; ++d) s += s_sumnf[d] * W1[d * 64 + tid];
        s_hidden[tid] = fmaxf(s, 0.0f);
    }
    __syncthreads();
    if (tid < 10) {
        float s = b2[tid];
        #pragma unroll 8
        for (int j = 0; j < 64; ++j) s += s_hidden[j] * W2[j * 10 + tid];
        out_qagg[(size_t)g * 10 + tid] = s_qsum[tid] + s;
    }
}

// ---------------------------------------------------------------------------
// d_out layout (concatenated flat, return order):
//   q_agg      [2048,10]      off 0        len 20480
//   ally_ws    [131072,10]    off 20480    len 1310720
//   wf         [2048,10,64]   off 1331200  len 1310720
//   ally_normed[131072,10]    off 2641920  len 1310720
//   normed_full[262144,10]    off 3952640  len 2621440
// ---------------------------------------------------------------------------
extern "C" void kernel_launch(void* const* d_in, const int* in_sizes, int n_in,
                              void* d_out, int out_size, void* d_ws, size_t ws_size,
                              hipStream_t stream) {
    (void)in_sizes; (void)n_in; (void)out_size; (void)d_ws; (void)ws_size;

    const float* node_feature = (const float*)d_in[0];
    const float* qs           = (const float*)d_in[1];
    const float* Ww           = (const float*)d_in[2];
    const float* bw           = (const float*)d_in[3];
    const float* W1           = (const float*)d_in[4];
    const float* b1           = (const float*)d_in[5];
    const float* W2           = (const float*)d_in[6];
    const float* b2           = (const float*)d_in[7];
    const int*   ally_indices = (const int*)d_in[8];
    // d_in[9] node_graph_ids: implicit (node // 128), unused.

    float* out        = (float*)d_out;
    float* out_qagg   = out;
    float* out_ws     = out + 20480;
    float* out_wf     = out + 1331200;
    float* out_normed = out + 2641920;
    float* out_full   = out + 3952640;

    // normed_full has zeros at non-ally rows; clear that slice first.
    qmixer_zero_kernel<<<2048, 256, 0, stream>>>(out_full, 2621440);

    qmixer_graph_kernel<<<2048, 128, 0, stream>>>(
        node_feature, qs, Ww, bw, W1, b1, W2, b2, ally_indices,
        out_qagg, out_ws, out_wf, out_normed, out_full);
}